// RDecoder_26817775796545
// MI455X (gfx1250) — compile-verified
//
#include <hip/hip_runtime.h>
#include <math.h>

typedef __attribute__((ext_vector_type(2))) float v2f;
typedef __attribute__((ext_vector_type(8))) float v8f;

#define T_STEPS 2048
#define BATCH   64
#define HID     512
#define CIN     64
#define NOUT    64

#define NWG_REC     32            // strips per layer (512 hidden / 16)
#define NWG_TOTAL   (2 * NWG_REC) // fused kernel: layer0 WGs + layer1 WGs
#define THREADS_REC 256           // 8 wave32s
#define THREADS_HEAD 256

// D = A(16x4 f32) * B(4x16 f32) + C(16x16 f32), full fp32 matrix core op
__device__ __forceinline__ v8f wmma4(v2f a, v2f b, v8f c) {
  return __builtin_amdgcn_wmma_f32_16x16x4_f32(false, a, false, b, (short)0, c,
                                               false, false);
}

// ---------------------------------------------------------------------------
// One GRU layer role inside the fused persistent kernel.
//   L0 (IS_L1=false): xin = input[B,CIN], ybase = 2-slot ring [2][B][HID]
//                     step t = phase p;  gi is time-invariant (precomputed).
//   L1 (IS_L1=true) : xin = ring (reads ys0[t] at slot t&1), ybase = ys1[T,B,HID]
//                     step t = phase p-1 (1-step pipeline lag).
// 8 waves: L0 -> wave pairs split K; L1 -> waves 0-3 h*W_hh, waves 4-7
// ys0[t]*W_ih; partials merged via LDS. One device barrier per phase.
// ---------------------------------------------------------------------------
template <bool IS_L1>
__device__ __forceinline__ void
run_layer(float* smem,
          const float* __restrict__ xin, const float* __restrict__ h0,
          const float* __restrict__ W_ih, const float* __restrict__ W_hh,
          const float* __restrict__ b_ih, const float* __restrict__ b_hh,
          float* __restrict__ ybase, unsigned* __restrict__ bar, int strip)
{
  float* whh  = smem;                                   // [48][HID]
  float* aux  = smem + 48 * HID;                        // L1: wih[48][HID]; L0: gi[48][BATCH]
  float* xfer = IS_L1 ? (smem + 2 * 48 * HID)           // [4][3][8][32]
                      : (smem + 48 * HID + 48 * BATCH);

  const int tid   = threadIdx.x;
  const int lane  = tid & 31;
  const int wv    = tid >> 5;          // 0..7
  const int mn    = lane & 15;         // N (or M for A-load)
  const int kh    = (lane >> 4) << 1;  // K sub-offset {0,2}
  const int hbase = strip << 4;        // hidden strip base
  const size_t BH = (size_t)BATCH * HID;

  // ---- stage W_hh strip rows {g*512+hbase+j} -> whh[(g*16+j)][k] ----
  for (int idx = tid; idx < 48 * HID; idx += THREADS_REC) {
    int lr = idx >> 9, k = idx & (HID - 1);
    int gate = lr >> 4, j = lr & 15;
    whh[idx] = W_hh[(size_t)(gate * HID + hbase + j) * HID + k];
  }
  if (IS_L1) {
    for (int idx = tid; idx < 48 * HID; idx += THREADS_REC) {
      int lr = idx >> 9, k = idx & (HID - 1);
      int gate = lr >> 4, j = lr & 15;
      aux[idx] = W_ih[(size_t)(gate * HID + hbase + j) * HID + k];
    }
  } else {
    // gi0 constant over time: gi0[b][col] = input[b,:].W_ih[col,:] + b_ih[col]
    for (int idx = tid; idx < 48 * BATCH; idx += THREADS_REC) {
      int lr = idx >> 6, b = idx & (BATCH - 1);
      int gate = lr >> 4, j = lr & 15;
      int col = gate * HID + hbase + j;
      float s = b_ih[col];
      for (int c = 0; c < CIN; ++c)
        s += xin[b * CIN + c] * W_ih[(size_t)col * CIN + c];
      aux[idx] = s;
    }
  }
  __syncthreads();

  const int  mt  = wv & 3;             // m-tile (16 batch rows)
  const int  m0  = mt << 4;
  const bool hiw = (wv >= 4);
  const int  hid_col = hbase + mn;

  const float bhr = b_hh[hid_col];
  const float bhz = b_hh[HID + hid_col];
  const float bhn = b_hh[2 * HID + hid_col];
  float bir = 0.f, biz = 0.f, bin_ = 0.f;
  if (IS_L1) {
    bir  = b_ih[hid_col];
    biz  = b_ih[HID + hid_col];
    bin_ = b_ih[2 * HID + hid_col];
  }

  // B-operand bases (LDS): hi-waves of L1 use W_ih, everyone else W_hh
  const float* wb = (IS_L1 && hiw) ? aux : whh;
  const float* br = wb + (0 * 16 + mn) * HID + kh;
  const float* bz = wb + (1 * 16 + mn) * HID + kh;
  const float* bn = wb + (2 * 16 + mn) * HID + kh;

  int k0, k1;
  if (IS_L1) { k0 = 0; k1 = HID; }                             // both GEMMs full K
  else       { k0 = hiw ? (HID / 2) : 0; k1 = k0 + HID / 2; }  // split K

  float* xf = xfer + (mt * 3) * 256;   // hi-wave writes, partner lo-wave reads

  for (int p = 0; p <= T_STEPS; ++p) {
    const int  t      = IS_L1 ? (p - 1) : p;
    const bool active = ((unsigned)t < (unsigned)T_STEPS);

    v8f ar = {}, az = {}, an = {};
    const float* hprev = h0;

    if (active) {
      if (t != 0)
        hprev = IS_L1 ? (ybase + (size_t)(t - 1) * BH)
                      : (ybase + (size_t)((t - 1) & 1) * BH);
      const float* arow;
      if (IS_L1 && hiw)   // gi GEMM A-operand: ys0[t] from ring slot t&1
        arow = xin + (size_t)(t & 1) * BH + (size_t)(m0 + mn) * HID + kh;
      else
        arow = hprev + (size_t)(m0 + mn) * HID + kh;

#pragma unroll 8
      for (int k = k0; k < k1; k += 4) {
        v2f a = *(const v2f*)(arow + k);
        ar = wmma4(a, *(const v2f*)(br + k), ar);
        az = wmma4(a, *(const v2f*)(bz + k), az);
        an = wmma4(a, *(const v2f*)(bn + k), an);
      }

      if (hiw) {
#pragma unroll
        for (int v = 0; v < 8; ++v) {
          xf[0 * 256 + v * 32 + lane] = ar[v];
          xf[1 * 256 + v * 32 + lane] = az[v];
          xf[2 * 256 + v * 32 + lane] = an[v];
        }
      }
    }
    __syncthreads();

    if (active && !hiw) {
      float* yout = IS_L1 ? (ybase + (size_t)t * BH)
                          : (ybase + (size_t)(t & 1) * BH);
#pragma unroll
      for (int v = 0; v < 8; ++v) {
        float xr = xf[0 * 256 + v * 32 + lane];
        float xz = xf[1 * 256 + v * 32 + lane];
        float xn = xf[2 * 256 + v * 32 + lane];
        int b = m0 + v + ((lane >> 4) << 3);
        float gir, giz, gin, ghr, ghz, ghn;
        if (IS_L1) {
          ghr = ar[v]; ghz = az[v]; ghn = an[v];
          gir = xr + bir; giz = xz + biz; gin = xn + bin_;
        } else {
          ghr = ar[v] + xr; ghz = az[v] + xz; ghn = an[v] + xn;
          gir = aux[(0 * 16 + mn) * BATCH + b];
          giz = aux[(1 * 16 + mn) * BATCH + b];
          gin = aux[(2 * 16 + mn) * BATCH + b];
        }
        float r  = 1.f / (1.f + expf(-(gir + ghr + bhr)));
        float z  = 1.f / (1.f + expf(-(giz + ghz + bhz)));
        float nn = tanhf(gin + r * (ghn + bhn));
        float hp = hprev[(size_t)b * HID + hid_col];
        yout[(size_t)b * HID + hid_col] = (1.f - z) * nn + z * hp;
      }
    }

    // ---- device-wide phase barrier (monotonic counter over 64 WGs) ----
    __threadfence();
    __syncthreads();
    if (tid == 0) {
      atomicAdd(bar, 1u);
      unsigned tgt = (unsigned)(p + 1) * NWG_TOTAL;
      while (__hip_atomic_load(bar, __ATOMIC_RELAXED, __HIP_MEMORY_SCOPE_AGENT) < tgt) {
        __builtin_amdgcn_s_sleep(2);
      }
    }
    __syncthreads();
    __threadfence();   // invalidate L0 caches so next phase sees remote writes
  }
}

__global__ void __launch_bounds__(THREADS_REC, 1)
gru_fused_kernel(const float* __restrict__ input, const float* __restrict__ hidden,
                 const float* __restrict__ W_ih0, const float* __restrict__ W_hh0,
                 const float* __restrict__ b_ih0, const float* __restrict__ b_hh0,
                 const float* __restrict__ W_ih1, const float* __restrict__ W_hh1,
                 const float* __restrict__ b_ih1, const float* __restrict__ b_hh1,
                 float* __restrict__ ring,  // [2][B][HID] ys0 ring
                 float* __restrict__ ys1,   // [T][B][HID]
                 unsigned* __restrict__ bar)
{
  extern __shared__ float smem[];
  if (blockIdx.x < NWG_REC)
    run_layer<false>(smem, input, hidden, W_ih0, W_hh0, b_ih0, b_hh0,
                     ring, bar, (int)blockIdx.x);
  else
    run_layer<true>(smem, ring, hidden + BATCH * HID, W_ih1, W_hh1, b_ih1, b_hh1,
                    ys1, bar, (int)blockIdx.x - NWG_REC);
}

// ---------------------------------------------------------------------------
// Head: out[b,t,o] = ys1[t,b,:] . W_lin[o,:] + b_lin[o]
// [T*B,512] x [512,64] fp32 WMMA GEMM; 8 waves/WG, 1 m-tile x 4 n-tiles/wave.
// ---------------------------------------------------------------------------
__global__ void __launch_bounds__(THREADS_HEAD, 1)
head_kernel(const float* __restrict__ ys1, const float* __restrict__ W_lin,
            const float* __restrict__ b_lin, float* __restrict__ out)
{
  const int lane = threadIdx.x & 31;
  const int wv   = threadIdx.x >> 5;
  const int mn   = lane & 15;
  const int kh   = (lane >> 4) << 1;
  const size_t r0 = ((size_t)blockIdx.x * 8 + wv) << 4;

  const float* arow = ys1 + (r0 + mn) * HID + kh;
  const float* w0 = W_lin + (size_t)(0 * 16 + mn) * HID + kh;
  const float* w1 = W_lin + (size_t)(1 * 16 + mn) * HID + kh;
  const float* w2 = W_lin + (size_t)(2 * 16 + mn) * HID + kh;
  const float* w3 = W_lin + (size_t)(3 * 16 + mn) * HID + kh;

  v8f a0 = {}, a1 = {}, a2 = {}, a3 = {};
#pragma unroll 4
  for (int k = 0; k < HID; k += 4) {
    v2f a = *(const v2f*)(arow + k);
    a0 = wmma4(a, *(const v2f*)(w0 + k), a0);
    a1 = wmma4(a, *(const v2f*)(w1 + k), a1);
    a2 = wmma4(a, *(const v2f*)(w2 + k), a2);
    a3 = wmma4(a, *(const v2f*)(w3 + k), a3);
  }
  const float bl0 = b_lin[mn],      bl1 = b_lin[16 + mn];
  const float bl2 = b_lin[32 + mn], bl3 = b_lin[48 + mn];
#pragma unroll
  for (int v = 0; v < 8; ++v) {
    size_t r = r0 + v + ((lane >> 4) << 3);  // flat row = t*64 + b
    int tt = (int)(r >> 6);
    int bb = (int)(r & 63);
    float* o = out + ((size_t)bb * T_STEPS + tt) * NOUT;
    o[mn]      = a0[v] + bl0;
    o[16 + mn] = a1[v] + bl1;
    o[32 + mn] = a2[v] + bl2;
    o[48 + mn] = a3[v] + bl3;
  }
}

// ---------------------------------------------------------------------------
extern "C" void kernel_launch(void* const* d_in, const int* in_sizes, int n_in,
                              void* d_out, int out_size, void* d_ws, size_t ws_size,
                              hipStream_t stream)
{
  (void)in_sizes; (void)n_in; (void)out_size;
  const float* input  = (const float*)d_in[0];
  const float* hidden = (const float*)d_in[1];   // [2,B,HID]
  const float* W_ih0  = (const float*)d_in[2];
  const float* W_hh0  = (const float*)d_in[3];
  const float* b_ih0  = (const float*)d_in[4];
  const float* b_hh0  = (const float*)d_in[5];
  const float* W_ih1  = (const float*)d_in[6];
  const float* W_hh1  = (const float*)d_in[7];
  const float* b_ih1  = (const float*)d_in[8];
  const float* b_hh1  = (const float*)d_in[9];
  const float* W_lin  = (const float*)d_in[10];
  const float* b_lin  = (const float*)d_in[11];
  float* out = (float*)d_out;

  const size_t BH       = (size_t)BATCH * HID;
  const size_t ys_elems = (size_t)T_STEPS * BH;
  float*    ys1  = (float*)d_ws;          // [T][B][HID]
  float*    ring = ys1 + ys_elems;        // [2][B][HID] ys0 ring
  unsigned* bars = (unsigned*)(ring + 2 * BH);
  if (ws_size < (ys_elems + 2 * BH) * sizeof(float) + 2 * sizeof(unsigned)) return;

  hipMemsetAsync(bars, 0, 2 * sizeof(unsigned), stream);

  const int shmem_rec = (2 * 48 * HID + 4 * 3 * 256) * (int)sizeof(float);
  (void)hipFuncSetAttribute(reinterpret_cast<const void*>(&gru_fused_kernel),
                            hipFuncAttributeMaxDynamicSharedMemorySize, shmem_rec);

  gru_fused_kernel<<<NWG_TOTAL, THREADS_REC, shmem_rec, stream>>>(
      input, hidden, W_ih0, W_hh0, b_ih0, b_hh0,
      W_ih1, W_hh1, b_ih1, b_hh1, ring, ys1, bars);

  head_kernel<<<(T_STEPS * BATCH / 16) / 8, THREADS_HEAD, 0, stream>>>(
      ys1, W_lin, b_lin, out);

  float* hout = out + (size_t)BATCH * T_STEPS * NOUT;
  // h_out[0] = ys0[T-1] (ring slot (T-1)&1), h_out[1] = ys1[T-1]
  hipMemcpyAsync(hout, ring + (size_t)((T_STEPS - 1) & 1) * BH,
                 BH * sizeof(float), hipMemcpyDeviceToDevice, stream);
  hipMemcpyAsync(hout + BH, ys1 + (size_t)(T_STEPS - 1) * BH,
                 BH * sizeof(float), hipMemcpyDeviceToDevice, stream);
}